// MaxPooling_1125281432061
// MI455X (gfx1250) — compile-verified
//
#include <hip/hip_runtime.h>
#include <hip/hip_bf16.h>
#include <math.h>

// ---------------- configuration ----------------
#define CCH 32                       // channels; == wave32 lane count (perfect fit)
#define BLOCK_THREADS 256            // 8 waves per block on CDNA5 wave32
#define WAVES_PER_BLOCK (BLOCK_THREADS / 32)
#define ROWS_PER_WAVE 64
#define BLOCK_ROWS (WAVES_PER_BLOCK * ROWS_PER_WAVE)   // 512 rows per block

// gfx1250 async global->LDS path, guarded so the file compiles on any toolchain.
#if __has_builtin(__builtin_amdgcn_global_load_async_to_lds_b32) && \
    __has_builtin(__builtin_amdgcn_s_wait_asynccnt)
#define USE_ASYNC_LDS 1
#else
#define USE_ASYNC_LDS 0
#endif

typedef __attribute__((address_space(1))) int global_int;
typedef __attribute__((address_space(3))) int lds_int;

__device__ __forceinline__ void atomic_fmax_f32(float* p, float v) {
    // Expect global_atomic_max_num_f32 on gfx1250.
    (void)__hip_atomic_fetch_max(p, v, __ATOMIC_RELAXED, __HIP_MEMORY_SCOPE_AGENT);
}

// ---------------- codegen probe (NEVER launched) ----------------
// Placed first in the file so the disasm snippet shows exactly how the
// async-to-LDS copy, s_wait_asynccnt, and the float max atomic lower.
__global__ void __launch_bounds__(32) cdna5_codegen_probe(float* p, const int* g) {
    __shared__ int s[32];
#if USE_ASYNC_LDS
    __builtin_amdgcn_global_load_async_to_lds_b32(
        (global_int*)(g + threadIdx.x), (lds_int*)(&s[threadIdx.x]), 0, 0);
    __builtin_amdgcn_s_wait_asynccnt(0);
#else
    s[threadIdx.x] = g[threadIdx.x];
#endif
    __syncthreads();
    atomic_fmax_f32(p + threadIdx.x, (float)s[threadIdx.x]);
}

// ---------------- kernel: sorted-segment max with run-tracking ----------------
__global__ void __launch_bounds__(BLOCK_THREADS)
seg_max_pool_kernel(const float* __restrict__ feat,
                    const int*   __restrict__ ids,
                    float*       __restrict__ out,
                    int n) {
    __shared__ int s_ids[BLOCK_ROWS];

    const int lane      = threadIdx.x & 31;
    const int wave      = threadIdx.x >> 5;
    const int blockBase = blockIdx.x * BLOCK_ROWS;
    const bool fullBlk  = (blockBase + BLOCK_ROWS) <= n;

    // Stage this block's segment ids into LDS. BLOCK_ROWS == 2*BLOCK_THREADS,
    // so exactly two straight-line guarded copies (no masked loop).
    {
        const int i0 = threadIdx.x;
        const int i1 = threadIdx.x + BLOCK_THREADS;
        const int r0 = blockBase + i0;
        const int r1 = blockBase + i1;
#if USE_ASYNC_LDS
        if (r0 < n) __builtin_amdgcn_global_load_async_to_lds_b32(
                        (global_int*)(ids + r0), (lds_int*)(&s_ids[i0]), 0, 0);
        else        s_ids[i0] = -1;
        if (r1 < n) __builtin_amdgcn_global_load_async_to_lds_b32(
                        (global_int*)(ids + r1), (lds_int*)(&s_ids[i1]), 0, 0);
        else        s_ids[i1] = -1;
        __builtin_amdgcn_s_wait_asynccnt(0);
#else
        s_ids[i0] = (r0 < n) ? ids[r0] : -1;
        s_ids[i1] = (r1 < n) ? ids[r1] : -1;
#endif
    }
    __syncthreads();

    // Each wave scans its 64 consecutive rows; lane == channel.
    const int rowBase = blockBase + wave * ROWS_PER_WAVE;
    const int sbase   = wave * ROWS_PER_WAVE;
    int   cur = -1;
    float m   = -INFINITY;

    if (fullBlk) {
        // Fast path: no per-row bounds checks.
#pragma unroll 4
        for (int i = 0; i < ROWS_PER_WAVE; ++i) {
            const int r = rowBase + i;
            if ((i & 7) == 0) {
                __builtin_prefetch(feat + (r + 16) * CCH + lane, 0, 1);
            }
            const int   seg = s_ids[sbase + i];        // wave-uniform
            const float v   = feat[r * CCH + lane];    // 128B coalesced per wave
            if (seg != cur) {
                if (cur >= 0) atomic_fmax_f32(&out[cur * CCH + lane], m);
                cur = seg;
                m   = v;
            } else {
                m = fmaxf(m, v);
            }
        }
    } else {
        // Guarded tail path (only the final block).
        for (int i = 0; i < ROWS_PER_WAVE; ++i) {
            const int r = rowBase + i;
            if (r >= n) break;
            const int   seg = s_ids[sbase + i];
            const float v   = feat[r * CCH + lane];
            if (seg != cur) {
                if (cur >= 0) atomic_fmax_f32(&out[cur * CCH + lane], m);
                cur = seg;
                m   = v;
            } else {
                m = fmaxf(m, v);
            }
        }
    }
    if (cur >= 0) atomic_fmax_f32(&out[cur * CCH + lane], m);
}

// ---------------- kernel: init out to -inf ----------------
__global__ void seg_init_kernel(float4* __restrict__ out4, int n4) {
    int i = blockIdx.x * blockDim.x + threadIdx.x;
    const float ninf = -INFINITY;
    float4 v = make_float4(ninf, ninf, ninf, ninf);
    if (i < n4) out4[i] = v;
}

// ---------------- kernel: replace +/-inf with 0 (empty parents) ----------------
__global__ void seg_finalize_kernel(float4* __restrict__ out4, int n4) {
    int i = blockIdx.x * blockDim.x + threadIdx.x;
    if (i >= n4) return;
    float4 v = out4[i];
    v.x = (fabsf(v.x) == INFINITY) ? 0.0f : v.x;
    v.y = (fabsf(v.y) == INFINITY) ? 0.0f : v.y;
    v.z = (fabsf(v.z) == INFINITY) ? 0.0f : v.z;
    v.w = (fabsf(v.w) == INFINITY) ? 0.0f : v.w;
    out4[i] = v;
}

extern "C" void kernel_launch(void* const* d_in, const int* in_sizes, int n_in,
                              void* d_out, int out_size, void* d_ws, size_t ws_size,
                              hipStream_t stream) {
    const float* feat = (const float*)d_in[0];   // [N, 32] float32
    const int*   ids  = (const int*)d_in[1];     // [N] int32, sorted
    (void)d_ws; (void)ws_size; (void)n_in;

    const int n  = in_sizes[0] / CCH;            // N rows
    float* out   = (float*)d_out;                // [M, 32] float32
    const int mc = out_size;                     // M * 32
    const int n4 = mc / 4;

    // 1) init to -inf
    seg_init_kernel<<<(n4 + 255) / 256, 256, 0, stream>>>((float4*)out, n4);

    // 2) segment max (run-tracking + L2 fmax atomics)
    const int grid = (n + BLOCK_ROWS - 1) / BLOCK_ROWS;
    seg_max_pool_kernel<<<grid, BLOCK_THREADS, 0, stream>>>(feat, ids, out, n);

    // 3) -inf -> 0 for empty parents
    seg_finalize_kernel<<<(n4 + 255) / 256, 256, 0, stream>>>((float4*)out, n4);
}